// Protonet_71150428226028
// MI455X (gfx1250) — compile-verified
//
#include <hip/hip_runtime.h>
#include <math.h>
#include <stdint.h>

typedef __attribute__((ext_vector_type(2))) float v2f;
typedef __attribute__((ext_vector_type(8))) float v8f;

#define N_CLASS   128
#define N_SUPPORT 32
#define N_QUERY   128
#define IN_DIM    2048
#define Z_DIM     1024
#define NS_ROWS   (N_CLASS * N_SUPPORT)   // 4096
#define NQ_ROWS   (N_CLASS * N_QUERY)     // 16384
#define M_TOTAL   (NS_ROWS + NQ_ROWS)     // 20480
#define N_PROTO   (N_CLASS * 2)           // 256
#define APAD      20                      // padded A row stride (80B: 16B-aligned, bank-conflict-free)

// ---------------------------------------------------------------------------
// CDNA5 async global->LDS copy (ASYNCcnt-tracked), 16 bytes per lane.
// LDS address from the vdst VGPR; global address is SGPR base plus VGPR offset (GVS mode)
// ---------------------------------------------------------------------------
__device__ __forceinline__ void async_cp16(uint32_t lds_addr, const void* sbase, uint32_t goff) {
    asm volatile("global_load_async_to_lds_b128 %0, %1, %2"
                 :: "v"(lds_addr), "v"(goff), "s"(sbase)
                 : "memory");
}

template <int N>
__device__ __forceinline__ void wait_asynccnt() {
    asm volatile("s_wait_asynccnt %0" :: "n"(N) : "memory");
}

__device__ __forceinline__ uint32_t lds_addr_of(const void* p) {
    // generic (flat) address of a __shared__ object: addr[31:0] is the LDS offset
    return (uint32_t)(uintptr_t)p;
}

// ---------------------------------------------------------------------------
// GEMM1: z[20480 x 1024] = concat(xs, xq)[20480 x 2048] @ W[2048 x 1024]
// Block tile 128x128, BK=16, double-buffered async LDS staging,
// 8 waves (2M x 4N), wave tile 64x32 (4x2 wmma), fp32 WMMA 16x16x4.
// ---------------------------------------------------------------------------
__global__ __launch_bounds__(256) void gemm1_kernel(const float* __restrict__ xs,
                                                    const float* __restrict__ xq,
                                                    const float* __restrict__ W,
                                                    float* __restrict__ z) {
    const int nblk = blockIdx.x;          // 0..7   (N)
    const int mblk = blockIdx.y;          // 0..159 (M) -> same x rows adjacent in schedule
    const int m0 = mblk * 128;
    const int n0 = nblk * 128;

    const float* src = (m0 < NS_ROWS) ? (xs + (size_t)m0 * IN_DIM)
                                      : (xq + (size_t)(m0 - NS_ROWS) * IN_DIM);

    __shared__ __attribute__((aligned(16))) float Am[2][128][APAD]; // row-major A tile
    __shared__ __attribute__((aligned(16))) float Bs[2][16][128];   // row-major B tile

    const int tid  = threadIdx.x;
    const int lane = tid & 31;
    const int wave = tid >> 5;
    const int hi   = lane >> 4;           // which K pair this half-wave holds
    const int lm   = lane & 15;
    const int wm   = wave & 1;            // 64 rows
    const int wn   = wave >> 1;           // 32 cols

    const v8f zerov = {0.f,0.f,0.f,0.f,0.f,0.f,0.f,0.f};
    v8f acc[4][2];
#pragma unroll
    for (int i = 0; i < 4; ++i)
#pragma unroll
        for (int j = 0; j < 2; ++j) acc[i][j] = zerov;

    // async staging thread mapping
    const int ar = tid >> 2;              // 0..63  (A row; +64 for 2nd issue)
    const int ac = (tid & 3) * 4;         // 0,4,8,12 (A 16B chunk)
    const int bk = tid >> 5;              // 0..7   (B row; +8 for 2nd issue)
    const int bn = (tid & 31) * 4;        // 0..124 (B 16B chunk)

    const uint32_t amb[2] = { lds_addr_of(&Am[0][0][0]), lds_addr_of(&Am[1][0][0]) };
    const uint32_t bsb[2] = { lds_addr_of(&Bs[0][0][0]), lds_addr_of(&Bs[1][0][0]) };

    auto stage = [&](int buf, int kb) {
        const uint32_t ldsA  = amb[buf] + (uint32_t)((ar * APAD + ac) * 4);
        const uint32_t goffA = (uint32_t)((ar * IN_DIM + kb + ac) * 4);
        async_cp16(ldsA,                    src, goffA);
        async_cp16(ldsA + 64u * APAD * 4u,  src, goffA + 64u * IN_DIM * 4u);
        const uint32_t ldsB  = bsb[buf] + (uint32_t)((bk * 128 + bn) * 4);
        const uint32_t goffB = (uint32_t)(((kb + bk) * Z_DIM + n0 + bn) * 4);
        async_cp16(ldsB,                    W, goffB);
        async_cp16(ldsB + 8u * 128u * 4u,   W, goffB + 8u * Z_DIM * 4u);
    };

    stage(0, 0);
    int buf = 0;
    for (int kb = 0; kb < IN_DIM; kb += 16, buf ^= 1) {
        if (kb + 16 < IN_DIM) {
            stage(buf ^ 1, kb + 16);     // prefetch next tile into other buffer
            wait_asynccnt<4>();          // current tile's 4 copies have landed
        } else {
            wait_asynccnt<0>();
        }
        __syncthreads();

        const float (*A)[APAD] = Am[buf];
        const float (*B)[128]  = Bs[buf];

#pragma unroll
        for (int kk = 0; kk < 16; kk += 4) {
            v2f af[4], bf[2];
#pragma unroll
            for (int i = 0; i < 4; ++i) {
                const int m = wm * 64 + i * 16 + lm;
                af[i] = *(const v2f*)&A[m][kk + 2 * hi];   // ISA A 16x4: V0={K0|K2}, V1={K1|K3}
            }
#pragma unroll
            for (int j = 0; j < 2; ++j) {
                const int n = wn * 32 + j * 16 + lm;
                bf[j].x = B[kk + 2 * hi + 0][n];           // B 4x16 mirrors A striping
                bf[j].y = B[kk + 2 * hi + 1][n];
            }
#pragma unroll
            for (int i = 0; i < 4; ++i)
#pragma unroll
                for (int j = 0; j < 2; ++j)
                    acc[i][j] = __builtin_amdgcn_wmma_f32_16x16x4_f32(
                        false, af[i], false, bf[j], (short)0, acc[i][j], false, false);
        }
        __syncthreads();   // reads done before next stage overwrites this buffer
    }

    // epilogue: C layout VGPR v, lane: M = v + 8*hi, N = lm
#pragma unroll
    for (int i = 0; i < 4; ++i)
#pragma unroll
        for (int j = 0; j < 2; ++j)
#pragma unroll
            for (int v = 0; v < 8; ++v) {
                const int row = m0 + wm * 64 + i * 16 + v + 8 * hi;
                const int col = n0 + wn * 32 + j * 16 + lm;
                z[(size_t)row * Z_DIM + col] = acc[i][j][v];
            }
}

// ---------------------------------------------------------------------------
// Per-class clustering (K=2): assignment to init centroids zs[c,0], zs[c,1],
// centroid means, write transposed prototypes protosT[d][256] + norms.
// ---------------------------------------------------------------------------
__global__ __launch_bounds__(256) void cluster_kernel(const float* __restrict__ z,
                                                      float* __restrict__ protosT,
                                                      float* __restrict__ pnorm) {
    const int c = blockIdx.x;
    const float* zc = z + (size_t)c * N_SUPPORT * Z_DIM;

    __shared__ int   sAssign[N_SUPPORT];
    __shared__ float sCount[2];
    __shared__ float red0[256], red1[256];

    const int tid  = threadIdx.x;
    const int wave = tid >> 5;
    const int lane = tid & 31;

    for (int si = 0; si < 4; ++si) {
        const int s = wave * 4 + si;
        const float* zrow = zc + (size_t)s * Z_DIM;
        float d0 = 0.f, d1 = 0.f;
        for (int d = lane; d < Z_DIM; d += 32) {
            const float v  = zrow[d];
            const float t0 = v - zc[d];
            const float t1 = v - zc[Z_DIM + d];
            d0 += t0 * t0;
            d1 += t1 * t1;
        }
#pragma unroll
        for (int off = 16; off > 0; off >>= 1) {
            d0 += __shfl_down(d0, off, 32);
            d1 += __shfl_down(d1, off, 32);
        }
        if (lane == 0) sAssign[s] = (d1 < d0) ? 1 : 0;   // argmin, first-index on tie
    }
    __syncthreads();

    if (tid == 0) {
        float c0 = 0.f, c1 = 0.f;
        for (int s = 0; s < N_SUPPORT; ++s) { if (sAssign[s]) c1 += 1.f; else c0 += 1.f; }
        sCount[0] = c0; sCount[1] = c1;
    }
    __syncthreads();

    const float cnt0 = sCount[0], cnt1 = sCount[1];
    float pn0 = 0.f, pn1 = 0.f;
    for (int d = tid; d < Z_DIM; d += 256) {
        float s0 = 0.f, s1 = 0.f;
        for (int s = 0; s < N_SUPPORT; ++s) {
            const float v = zc[(size_t)s * Z_DIM + d];
            if (sAssign[s]) s1 += v; else s0 += v;
        }
        const float cent0 = (cnt0 > 0.f) ? s0 / fmaxf(cnt0, 1.f) : zc[d];
        const float cent1 = (cnt1 > 0.f) ? s1 / fmaxf(cnt1, 1.f) : zc[Z_DIM + d];
        protosT[(size_t)d * N_PROTO + 2 * c + 0] = cent0;
        protosT[(size_t)d * N_PROTO + 2 * c + 1] = cent1;
        pn0 += cent0 * cent0;
        pn1 += cent1 * cent1;
    }
    red0[tid] = pn0; red1[tid] = pn1;
    __syncthreads();
#pragma unroll
    for (int off = 128; off > 0; off >>= 1) {
        if (tid < off) { red0[tid] += red0[tid + off]; red1[tid] += red1[tid + off]; }
        __syncthreads();
    }
    if (tid == 0) {
        pnorm[2 * c + 0] = red0[0];
        pnorm[2 * c + 1] = red1[0];
    }
}

// ---------------------------------------------------------------------------
// ||zq_row||^2, one wave (32 lanes) per row.
// ---------------------------------------------------------------------------
__global__ __launch_bounds__(256) void qnorm_kernel(const float* __restrict__ zq,
                                                    float* __restrict__ qn) {
    const int row  = blockIdx.x * 8 + (threadIdx.x >> 5);
    const int lane = threadIdx.x & 31;
    const float* zr = zq + (size_t)row * Z_DIM;
    float s = 0.f;
    for (int d = lane; d < Z_DIM; d += 32) { const float v = zr[d]; s += v * v; }
#pragma unroll
    for (int off = 16; off > 0; off >>= 1) s += __shfl_down(s, off, 32);
    if (lane == 0) qn[row] = s;
}

// ---------------------------------------------------------------------------
// GEMM2 + distance epilogue: dot = zq @ protosT (16384x1024 @ 1024x256),
// d2 = qn + pn - 2*dot, clamp, min over K=2 pair via lane-xor shuffle.
// Block tile 128x256, double-buffered async staging, 8 waves, wave 64x64.
// ---------------------------------------------------------------------------
__global__ __launch_bounds__(256) void gemm2_kernel(const float* __restrict__ zq,
                                                    const float* __restrict__ protosT,
                                                    const float* __restrict__ qn,
                                                    const float* __restrict__ pn,
                                                    float* __restrict__ dists) {
    const int m0 = blockIdx.x * 128;

    __shared__ __attribute__((aligned(16))) float Am[2][128][APAD];
    __shared__ __attribute__((aligned(16))) float Bs[2][16][256];

    const int tid  = threadIdx.x;
    const int lane = tid & 31;
    const int wave = tid >> 5;
    const int hi   = lane >> 4;
    const int lm   = lane & 15;
    const int wm   = wave & 1;            // 64 rows
    const int wn   = wave >> 1;           // 64 cols

    const v8f zerov = {0.f,0.f,0.f,0.f,0.f,0.f,0.f,0.f};
    v8f acc[4][4];
#pragma unroll
    for (int i = 0; i < 4; ++i)
#pragma unroll
        for (int j = 0; j < 4; ++j) acc[i][j] = zerov;

    const int ar = tid >> 2;              // 0..63
    const int ac = (tid & 3) * 4;         // 0,4,8,12
    const int bk = tid >> 6;              // 0..3  (B row; +4,+8,+12 for other issues)
    const int bn = (tid & 63) * 4;        // 0..252

    const uint32_t amb[2] = { lds_addr_of(&Am[0][0][0]), lds_addr_of(&Am[1][0][0]) };
    const uint32_t bsb[2] = { lds_addr_of(&Bs[0][0][0]), lds_addr_of(&Bs[1][0][0]) };

    const float* srcA = zq + (size_t)m0 * Z_DIM;

    auto stage = [&](int buf, int kb) {
        const uint32_t ldsA  = amb[buf] + (uint32_t)((ar * APAD + ac) * 4);
        const uint32_t goffA = (uint32_t)((ar * Z_DIM + kb + ac) * 4);
        async_cp16(ldsA,                   srcA, goffA);
        async_cp16(ldsA + 64u * APAD * 4u, srcA, goffA + 64u * Z_DIM * 4u);
#pragma unroll
        for (int rr = 0; rr < 4; ++rr) {
            const int k = bk + rr * 4;
            const uint32_t ldsB  = bsb[buf] + (uint32_t)((k * 256 + bn) * 4);
            const uint32_t goffB = (uint32_t)(((kb + k) * N_PROTO + bn) * 4);
            async_cp16(ldsB, protosT, goffB);
        }
    };

    stage(0, 0);
    int buf = 0;
    for (int kb = 0; kb < Z_DIM; kb += 16, buf ^= 1) {
        if (kb + 16 < Z_DIM) {
            stage(buf ^ 1, kb + 16);
            wait_asynccnt<6>();          // 2 A + 4 B copies pending from the new stage
        } else {
            wait_asynccnt<0>();
        }
        __syncthreads();

        const float (*A)[APAD] = Am[buf];
        const float (*B)[256]  = Bs[buf];

#pragma unroll
        for (int kk = 0; kk < 16; kk += 4) {
            v2f af[4], bf[4];
#pragma unroll
            for (int i = 0; i < 4; ++i) {
                const int m = wm * 64 + i * 16 + lm;
                af[i] = *(const v2f*)&A[m][kk + 2 * hi];
            }
#pragma unroll
            for (int j = 0; j < 4; ++j) {
                const int n = wn * 64 + j * 16 + lm;
                bf[j].x = B[kk + 2 * hi + 0][n];
                bf[j].y = B[kk + 2 * hi + 1][n];
            }
#pragma unroll
            for (int i = 0; i < 4; ++i)
#pragma unroll
                for (int j = 0; j < 4; ++j)
                    acc[i][j] = __builtin_amdgcn_wmma_f32_16x16x4_f32(
                        false, af[i], false, bf[j], (short)0, acc[i][j], false, false);
        }
        __syncthreads();
    }

    // epilogue: d2 = qn + pn - 2*dot ; clamp ; min over proto pair (2c, 2c+1)
#pragma unroll
    for (int i = 0; i < 4; ++i)
#pragma unroll
        for (int j = 0; j < 4; ++j) {
            const int nb = wn * 64 + j * 16;
            const int n  = nb + lm;
            const float pnv = pn[n];
#pragma unroll
            for (int v = 0; v < 8; ++v) {
                const int row = m0 + wm * 64 + i * 16 + v + 8 * hi;
                float d2 = qn[row] + pnv - 2.f * acc[i][j][v];
                d2 = fmaxf(d2, 0.f);
                const float other = __shfl_xor(d2, 1, 32);  // pair: N even <-> N odd
                if ((lm & 1) == 0)
                    dists[(size_t)row * N_CLASS + (n >> 1)] = fminf(d2, other);
            }
        }
}

// ---------------------------------------------------------------------------
// Per-row: sum over classes, first-index argmax, log(p_true). 128 thr/block.
// ---------------------------------------------------------------------------
__global__ __launch_bounds__(128) void loss_rows_kernel(const float* __restrict__ dists,
                                                        float* __restrict__ pl,
                                                        float* __restrict__ pa) {
    const int r = blockIdx.x;
    const int c = threadIdx.x;

    __shared__ float ssum[128];
    __shared__ float smax[128];
    __shared__ int   sidx[128];

    const float v = dists[(size_t)r * N_CLASS + c];
    ssum[c] = v; smax[c] = v; sidx[c] = c;
    __syncthreads();
#pragma unroll
    for (int off = 64; off > 0; off >>= 1) {
        if (c < off) {
            ssum[c] += ssum[c + off];
            const float b  = smax[c + off];
            const int   bi = sidx[c + off];
            if (b > smax[c] || (b == smax[c] && bi < sidx[c])) { smax[c] = b; sidx[c] = bi; }
        }
        __syncthreads();
    }
    if (c == 0) {
        const int ctrue = r >> 7;   // r / N_QUERY
        const float p = dists[(size_t)r * N_CLASS + ctrue] / ssum[0];
        pl[r] = logf(p);
        pa[r] = (sidx[0] == ctrue) ? 1.f : 0.f;
    }
}

// ---------------------------------------------------------------------------
// Deterministic final reduction -> d_out = {loss, acc}
// ---------------------------------------------------------------------------
__global__ __launch_bounds__(256) void finalize_kernel(const float* __restrict__ pl,
                                                       const float* __restrict__ pa,
                                                       float* __restrict__ out) {
    __shared__ float s0[256], s1[256];
    const int t = threadIdx.x;
    float a = 0.f, b = 0.f;
    for (int i = t; i < NQ_ROWS; i += 256) { a += pl[i]; b += pa[i]; }
    s0[t] = a; s1[t] = b;
    __syncthreads();
#pragma unroll
    for (int off = 128; off > 0; off >>= 1) {
        if (t < off) { s0[t] += s0[t + off]; s1[t] += s1[t + off]; }
        __syncthreads();
    }
    if (t == 0) {
        out[0] = -s0[0] / (float)NQ_ROWS;
        out[1] =  s1[0] / (float)NQ_ROWS;
    }
}

// ---------------------------------------------------------------------------
extern "C" void kernel_launch(void* const* d_in, const int* in_sizes, int n_in,
                              void* d_out, int out_size, void* d_ws, size_t ws_size,
                              hipStream_t stream) {
    const float* xs = (const float*)d_in[0];   // 128*32*2048
    const float* xq = (const float*)d_in[1];   // 128*128*2048
    const float* W  = (const float*)d_in[2];   // 2048*1024
    float* out = (float*)d_out;

    float* ws      = (float*)d_ws;
    float* z       = ws;                                   // 20480*1024
    float* protosT = z       + (size_t)M_TOTAL * Z_DIM;    // 1024*256
    float* pnorm   = protosT + (size_t)Z_DIM * N_PROTO;    // 256
    float* qnorm   = pnorm   + N_PROTO;                    // 16384
    float* dists   = qnorm   + NQ_ROWS;                    // 16384*128
    float* pl      = dists   + (size_t)NQ_ROWS * N_CLASS;  // 16384
    float* pa      = pl      + NQ_ROWS;                    // 16384

    const float* zq = z + (size_t)NS_ROWS * Z_DIM;

    gemm1_kernel  <<<dim3(Z_DIM / 128, M_TOTAL / 128), 256, 0, stream>>>(xs, xq, W, z);
    cluster_kernel<<<N_CLASS, 256, 0, stream>>>(z, protosT, pnorm);
    qnorm_kernel  <<<NQ_ROWS / 8, 256, 0, stream>>>(zq, qnorm);
    gemm2_kernel  <<<NQ_ROWS / 128, 256, 0, stream>>>(zq, protosT, qnorm, pnorm, dists);
    loss_rows_kernel<<<NQ_ROWS, 128, 0, stream>>>(dists, pl, pa);
    finalize_kernel <<<1, 256, 0, stream>>>(pl, pa, out);
}